// RoIPooling_12068858102123
// MI455X (gfx1250) — compile-verified
//
#include <hip/hip_runtime.h>
#include <stdint.h>

// Problem constants (from reference setup)
#define BB 2
#define RR 48
#define HH 56
#define WW 56
#define CC 256
#define PH 7
#define PW 7
#define HALF_C 128           // channels handled per wave (one of two halves)
#define MAX_BIN 16           // max bin extent (actual <= 14 = 56/7 + 7 - 1)

// -------- CDNA5 TDM availability (device pass only) --------
#if defined(__gfx1250__) && __has_builtin(__builtin_amdgcn_tensor_load_to_lds) && __has_builtin(__builtin_amdgcn_s_wait_tensorcnt)
#define HAVE_TDM 1
#else
#define HAVE_TDM 0
#endif

#if HAVE_TDM
typedef unsigned int u32x4 __attribute__((ext_vector_type(4)));
typedef int          i32x4 __attribute__((ext_vector_type(4)));
typedef int          i32x8 __attribute__((ext_vector_type(8)));

// Issue one TDM transfer: a 2D tile (HALF_C contiguous floats per line,
// ncols lines, line stride CC floats) from global memory into LDS at lds_off.
// D# layout per CDNA5 ISA ch.8 (groups 0..3).
__device__ __forceinline__ void tdm_load_tile(uint32_t lds_off, const float* gsrc, int ncols) {
  uint64_t ga = (uint64_t)(uintptr_t)gsrc;
  u32x4 g0;
  g0[0] = 1u;                                            // count=1, is_restore=0, gather off
  g0[1] = lds_off;                                       // lds_addr (bytes)
  g0[2] = (uint32_t)ga;                                  // global_addr[31:0]
  g0[3] = ((uint32_t)(ga >> 32) & 0x01FFFFFFu)           // global_addr[56:32]
          | (2u << 30);                                  // type=2 ("image")
  i32x8 g1;
  g1[0] = 0x00020000;                                    // workgroup_mask=0, data_size=2 (4B)
  g1[1] = (int)((unsigned)HALF_C << 16);                 // tensor_dim0[15:0]=128 (atomic_bar_addr=0)
  g1[2] = (int)(((unsigned)ncols & 0xFFFFu) << 16);      // tensor_dim1[15:0]=ncols
  g1[3] = (int)((unsigned)HALF_C << 16);                 // tile_dim0=128 (tensor_dim1[31:16]=0)
  g1[4] = ncols;                                         // tile_dim1=ncols, tile_dim2=0 (unused)
  g1[5] = CC;                                            // tensor_dim0_stride[31:0] = 256 elements
  g1[6] = 0;                                             // stride hi / tensor_dim1_stride lo = 0
  g1[7] = 0;
  i32x4 g2 = {1, 1, 0, 0};                               // tensor_dim2=1, tensor_dim3=1, stride2=0, tile_dim3=0
  i32x4 g3 = {0, (int)(1u << 16), 0, 0};                 // dim3_stride=0, tensor_dim4=1, tile_dim4=0
#if __clang_major__ >= 23
  i32x8 g4 = {0, 0, 0, 0, 0, 0, 0, 0};
  __builtin_amdgcn_tensor_load_to_lds(g0, g1, g2, g3, g4, 0);
#else
  __builtin_amdgcn_tensor_load_to_lds(g0, g1, g2, g3, 0);
#endif
}
#endif // HAVE_TDM

__global__ __launch_bounds__(32)
void roi_maxpool_kernel(const float* __restrict__ fm,
                        const float* __restrict__ rois,
                        float* __restrict__ out) {
  // 3-D grid: x = (j, half) in [0,14), y = i in [0,7), z = b*48 + r in [0,96).
  // Cheap decode (single small division) so the first TDM issues ASAP.
  const int half = (int)blockIdx.x & 1;
  const int j    = (int)blockIdx.x >> 1;
  const int i    = (int)blockIdx.y;
  const int z    = (int)blockIdx.z;
  const int b    = z / RR;
  const int r    = z - b * RR;
  const int lane = (int)threadIdx.x;   // 0..31, each lane owns 4 channels (float4)

  // ROI -> bin window, exactly mirroring the reference float math + truncation.
  const float4 roi = *(const float4*)(rois + (size_t)(b * RR + r) * 4);
  const float rx = roi.x, rymax = roi.y, rw_f = roi.z, rh_f = roi.w;
  const int w_start = (int)((float)WW * rx);
  const int w_end   = (int)((float)WW * (rx + rw_f));
  const int h_start = (int)((float)HH * (1.0f - rymax));
  const int h_end   = (int)((float)HH * ((1.0f - rymax) + rh_f));
  const int rh = h_end - h_start;
  const int rw = w_end - w_start;
  const int h_step = rh / PH;
  const int w_step = rw / PW;
  const int y0 = h_start + i * h_step;
  const int y1 = h_start + ((i == PH - 1) ? rh : (i + 1) * h_step);
  const int x0 = w_start + j * w_step;
  const int x1 = w_start + ((j == PW - 1) ? rw : (j + 1) * w_step);
  const int nrows = y1 - y0;
  const int ncols = x1 - x0;

  // Dual accumulators break the serial v_max dependence across the 2x-unrolled loop.
  float4 acc0, acc1;
  acc0.x = -INFINITY; acc0.y = -INFINITY; acc0.z = -INFINITY; acc0.w = -INFINITY;
  acc1 = acc0;

  // Global address of the bin's first pixel, at this wave's channel half.
  const float* gbase = fm + ((size_t)(b * HH + y0) * WW + (size_t)x0) * CC + (size_t)half * HALF_C;

#if HAVE_TDM
  // Double-buffered TDM pipeline: DMA row k+1 into LDS while max-reducing row k.
  __shared__ float buf[2][MAX_BIN * HALF_C];   // 2 x 8KB -> ~20 blocks/WGP occupancy
  if (nrows > 0 && ncols > 0) {
    tdm_load_tile((uint32_t)(uintptr_t)&buf[0][0], gbase, ncols);
    for (int k = 0; k < nrows; ++k) {
      if (k + 1 < nrows) {
        // WAR guard: ensure prior ds reads of the buffer being refilled are done.
        asm volatile("s_wait_dscnt 0" ::: "memory");
        tdm_load_tile((uint32_t)(uintptr_t)&buf[(k + 1) & 1][0],
                      gbase + (size_t)(k + 1) * WW * CC, ncols);
        __builtin_amdgcn_s_wait_tensorcnt(1);   // TENSORcnt in-order: row k complete
      } else {
        __builtin_amdgcn_s_wait_tensorcnt(0);
      }
      asm volatile("" ::: "memory");            // compiler fence: LDS now holds row k
      const float4* rowbuf = (const float4*)&buf[k & 1][0];
      int p = 0;
      for (; p + 2 <= ncols; p += 2) {          // two ds_load_b128 in flight
        float4 v0 = rowbuf[(size_t)(p + 0) * (HALF_C / 4) + lane];
        float4 v1 = rowbuf[(size_t)(p + 1) * (HALF_C / 4) + lane];
        acc0.x = fmaxf(acc0.x, v0.x); acc0.y = fmaxf(acc0.y, v0.y);
        acc0.z = fmaxf(acc0.z, v0.z); acc0.w = fmaxf(acc0.w, v0.w);
        acc1.x = fmaxf(acc1.x, v1.x); acc1.y = fmaxf(acc1.y, v1.y);
        acc1.z = fmaxf(acc1.z, v1.z); acc1.w = fmaxf(acc1.w, v1.w);
      }
      if (p < ncols) {
        float4 v = rowbuf[(size_t)p * (HALF_C / 4) + lane];
        acc0.x = fmaxf(acc0.x, v.x); acc0.y = fmaxf(acc0.y, v.y);
        acc0.z = fmaxf(acc0.z, v.z); acc0.w = fmaxf(acc0.w, v.w);
      }
    }
  }
#else
  // Fallback: direct coalesced global_load_b128 reduction.
  for (int k = 0; k < nrows; ++k) {
    const float4* rowp = (const float4*)(gbase + (size_t)k * WW * CC);
    int p = 0;
    for (; p + 2 <= ncols; p += 2) {
      float4 v0 = rowp[(size_t)(p + 0) * (CC / 4) + lane];
      float4 v1 = rowp[(size_t)(p + 1) * (CC / 4) + lane];
      acc0.x = fmaxf(acc0.x, v0.x); acc0.y = fmaxf(acc0.y, v0.y);
      acc0.z = fmaxf(acc0.z, v0.z); acc0.w = fmaxf(acc0.w, v0.w);
      acc1.x = fmaxf(acc1.x, v1.x); acc1.y = fmaxf(acc1.y, v1.y);
      acc1.z = fmaxf(acc1.z, v1.z); acc1.w = fmaxf(acc1.w, v1.w);
    }
    if (p < ncols) {
      float4 v = rowp[(size_t)p * (CC / 4) + lane];
      acc0.x = fmaxf(acc0.x, v.x); acc0.y = fmaxf(acc0.y, v.y);
      acc0.z = fmaxf(acc0.z, v.z); acc0.w = fmaxf(acc0.w, v.w);
    }
  }
#endif

  // Merge dual accumulators.
  acc0.x = fmaxf(acc0.x, acc1.x);
  acc0.y = fmaxf(acc0.y, acc1.y);
  acc0.z = fmaxf(acc0.z, acc1.z);
  acc0.w = fmaxf(acc0.w, acc1.w);

  // out[b, r, i, j, half*128 + lane*4 .. +3] -- one global_store_b128 per lane.
  float* dst = out + (((size_t)(b * RR + r) * PH + i) * PW + j) * CC
                   + (size_t)half * HALF_C + (size_t)lane * 4;
  *(float4*)dst = acc0;
}

extern "C" void kernel_launch(void* const* d_in, const int* in_sizes, int n_in,
                              void* d_out, int out_size, void* d_ws, size_t ws_size,
                              hipStream_t stream) {
  const float* fm   = (const float*)d_in[0];   // [2,56,56,256] f32
  const float* rois = (const float*)d_in[1];   // [2,48,4] f32
  float* out = (float*)d_out;                  // [2,48,7,7,256] f32

  // grid: x = (j,half)=14, y = i=7, z = b*48+r=96  -> 9408 single-wave blocks
  roi_maxpool_kernel<<<dim3(PW * 2, PH, BB * RR), dim3(32), 0, stream>>>(fm, rois, out);
}